// MixtureLogistic256_23905787969961
// MI455X (gfx1250) — compile-verified
//
#include <hip/hip_runtime.h>
#include <hip/hip_bf16.h>

// Problem constants (match reference setup_inputs)
#define BB    32
#define CC    3
#define MM    10
#define HWD   16384              // 128*128
#define TPB   256                // 8 wave32s per block
#define PPT   4                  // pixels per thread (float4 streaming)
#define BPIX  (TPB * PPT)        // 1024 pixels per block

#define LOG2E_F      1.4426950408889634f   // log2(e)
#define LN2_F        0.6931471805599453f   // ln(2)
#define LOG2_LOG2E_F 0.5287663729448977f   // log2(log2(e))

typedef float v4f __attribute__((ext_vector_type(4)));
typedef __attribute__((address_space(3))) float lds_f32;

__device__ __forceinline__ float fast_rcp(float v) {
    return __builtin_amdgcn_rcpf(v);
}

// Native base-2 exp/log: single v_exp_f32 / v_log_f32, no ln2<->log2e scaling mul.
__device__ __forceinline__ float fast_exp2(float v) {
#if __has_builtin(__builtin_amdgcn_exp2f)
    return __builtin_amdgcn_exp2f(v);
#else
    return exp2f(v);
#endif
}

__device__ __forceinline__ float fast_log2(float v) {
#if __has_builtin(__builtin_amdgcn_logf)
    return __builtin_amdgcn_logf(v);      // base-2 log
#else
    return __log2f(v);
#endif
}

__device__ __forceinline__ float fast_tanh(float v) {
#if __has_builtin(__builtin_amdgcn_tanhf)
    return __builtin_amdgcn_tanhf(v);     // native v_tanh_f32 on gfx1250
#else
    float z = fminf(fmaxf(v, -15.0f), 15.0f);
    float t = fast_exp2((2.0f * LOG2E_F) * z);   // e^{2z}
    return (t - 1.0f) * fast_rcp(t + 1.0f);
#endif
}

// Discretized-logistic per-channel log-prob in LOG2 domain.
// isl2 = inv_stdv * log2(e); plus_l2/min_l2 are the boundary inputs * log2(e).
//   interior: log2(max(sig_p - sig_m, 1e-10))
//   low  (xc<=pix0):   log2 sigmoid(plus_in)   = min(plus_l2,0)  + log2(rp)
//   high (xc>=pix255): log2(1-sigmoid(min_in)) = min(-min_l2,0)  + log2(rm)
// Caller multiplies the summed result by ln(2) once at the end.
__device__ __forceinline__ float chan_logprob_l2(float xc, float mu, float isl2,
                                                 bool lo, bool hi) {
    const float inv255 = 1.0f / 255.0f;
    float centered = xc - mu;
    float plus_l2 = isl2 * (centered + inv255);
    float min_l2  = isl2 * (centered - inv255);

    float ep = fast_exp2(-fabsf(plus_l2));   // = exp(-|plus_in|), 1 instr (src mods)
    float em = fast_exp2(-fabsf(min_l2));    // = exp(-|min_in|),  1 instr
    float rp = fast_rcp(1.0f + ep);
    float rm = fast_rcp(1.0f + em);
    float sig_p = (plus_l2 >= 0.0f) ? rp : ep * rp;
    float sig_m = (min_l2  >= 0.0f) ? rm : em * rm;

    float arg = fmaxf(sig_p - sig_m, 1e-10f);
    float add = 0.0f;
    arg = lo ? rp : arg;  add = lo ? fminf(plus_l2,  0.0f) : add;
    arg = hi ? rm : arg;  add = hi ? fminf(-min_l2, 0.0f) : add;
    return add + fast_log2(arg);
}

__global__ __launch_bounds__(64) void zero_out_kernel(float* __restrict__ out) {
    if (threadIdx.x < BB) out[threadIdx.x] = 0.0f;
}

__global__ __launch_bounds__(TPB) void dmol_kernel(
    const float* __restrict__ x,
    const float* __restrict__ logit_probs,
    const float* __restrict__ mean,
    const float* __restrict__ log_var,
    const float* __restrict__ coeffs,
    float* __restrict__ out)
{
    __shared__ float s_logit[MM * BPIX];   // 40 KB logit tile
    __shared__ float s_warp[TPB / 32];

    const int  tid = threadIdx.x;
    const long p0  = (long)blockIdx.x * BPIX;    // first pixel of this block
    const int  b   = (int)(p0 / HWD);            // whole block shares b (1024 | HWD)
    const int  hw  = (int)(p0 % HWD) + tid * PPT;

    // ---------- async DMA: stage logit_probs[b, :, tile] into LDS (B128) ----------
    {
        const float* lg_base = logit_probs + (long)b * MM * HWD + hw;
        unsigned lds0 = (unsigned)(unsigned long long)(lds_f32*)(&s_logit[tid * PPT]);
        #pragma unroll
        for (int m = 0; m < MM; ++m) {
            unsigned long long gaddr = (unsigned long long)(const void*)(lg_base + (long)m * HWD);
            unsigned laddr = lds0 + (unsigned)(m * BPIX * 4);
            asm volatile("global_load_async_to_lds_b128 %0, %1, off"
                         :: "v"(laddr), "v"(gaddr)
                         : "memory");
        }
        asm volatile("s_wait_asynccnt 0x0" ::: "memory");
    }
    __syncthreads();

    // ---------- per-pixel inputs (b128 NT streaming) ----------
    const long xbase = (long)b * CC * HWD + hw;
    v4f xv[CC];
    #pragma unroll
    for (int c = 0; c < CC; ++c) {
        v4f t = __builtin_nontemporal_load((const v4f*)(x + xbase + (long)c * HWD));
        xv[c] = 2.0f * t - 1.0f;
    }

    // boundary masks are m-invariant: hoist out of the mixture loop
    bool lo[CC][PPT], hi[CC][PPT];
    #pragma unroll
    for (int c = 0; c < CC; ++c)
        #pragma unroll
        for (int j = 0; j < PPT; ++j) {
            lo[c][j] = (xv[c][j] <= -1.0f + 1.0f / 255.0f);
            hi[c][j] = (xv[c][j] >=  1.0f - 1.0f / 255.0f);
        }

    // ---------- log_softmax denominator over M (from LDS tile) ----------
    float lgmax[PPT], lgsum[PPT], lden[PPT];
    #pragma unroll
    for (int j = 0; j < PPT; ++j) { lgmax[j] = -INFINITY; lgsum[j] = 0.0f; }
    #pragma unroll
    for (int m = 0; m < MM; ++m) {
        v4f lg = *(const v4f*)&s_logit[m * BPIX + tid * PPT];
        #pragma unroll
        for (int j = 0; j < PPT; ++j) lgmax[j] = fmaxf(lgmax[j], lg[j]);
    }
    #pragma unroll
    for (int m = 0; m < MM; ++m) {
        v4f lg = *(const v4f*)&s_logit[m * BPIX + tid * PPT];
        #pragma unroll
        for (int j = 0; j < PPT; ++j)
            lgsum[j] += fast_exp2(LOG2E_F * (lg[j] - lgmax[j]));
    }
    #pragma unroll
    for (int j = 0; j < PPT; ++j)
        lden[j] = lgmax[j] + LN2_F * fast_log2(lgsum[j]);

    // ---------- main mixture loop (log2 domain) ----------
    const long tbase = (long)b * CC * MM * HWD + hw;   // + (c*MM + m)*HWD
    float lpv[MM][PPT];                                 // lp * log2(e)
    float lpmax[PPT];
    #pragma unroll
    for (int j = 0; j < PPT; ++j) lpmax[j] = -INFINITY;

    #pragma unroll
    for (int m = 0; m < MM; ++m) {
        v4f mu[CC], lv[CC], cf[CC];
        #pragma unroll
        for (int c = 0; c < CC; ++c) {
            const long o = tbase + (long)(c * MM + m) * HWD;
            mu[c] = __builtin_nontemporal_load((const v4f*)(mean    + o));
            lv[c] = __builtin_nontemporal_load((const v4f*)(log_var + o));
            cf[c] = __builtin_nontemporal_load((const v4f*)(coeffs  + o));
        }
        v4f lg = *(const v4f*)&s_logit[m * BPIX + tid * PPT];

        #pragma unroll
        for (int j = 0; j < PPT; ++j) {
            float l0 = fminf(fmaxf(lv[0][j], -8.0f), 1.0f);
            float l1 = fminf(fmaxf(lv[1][j], -8.0f), 1.0f);
            float l2 = fminf(fmaxf(lv[2][j], -8.0f), 1.0f);
            // isl2 = inv_stdv * log2(e) = 2^(-lv*log2e + log2(log2e)); fma + exp2
            float is0 = fast_exp2(fmaf(-LOG2E_F, l0, LOG2_LOG2E_F));
            float is1 = fast_exp2(fmaf(-LOG2E_F, l1, LOG2_LOG2E_F));
            float is2 = fast_exp2(fmaf(-LOG2E_F, l2, LOG2_LOG2E_F));

            float c0 = fast_tanh(cf[0][j]);
            float c1 = fast_tanh(cf[1][j]);
            float c2 = fast_tanh(cf[2][j]);

            float x0 = xv[0][j], x1 = xv[1][j];
            float m0a = mu[0][j];
            float m1a = mu[1][j] + c0 * x0;
            float m2a = mu[2][j] + c1 * x0 + c2 * x1;

            float ch = chan_logprob_l2(x0,       m0a, is0, lo[0][j], hi[0][j]);
            ch      += chan_logprob_l2(x1,       m1a, is1, lo[1][j], hi[1][j]);
            ch      += chan_logprob_l2(xv[2][j], m2a, is2, lo[2][j], hi[2][j]);

            // lp_l2 = (lg - lden)*log2e + sum of log2-domain channel terms
            float lp_l2 = fmaf(lg[j] - lden[j], LOG2E_F, ch);
            lpv[m][j] = lp_l2;
            lpmax[j]  = fmaxf(lpmax[j], lp_l2);
        }
    }

    // ---------- logsumexp over M (log2 domain), sum the thread's 4 pixels ----------
    float acc = 0.0f;
    #pragma unroll
    for (int j = 0; j < PPT; ++j) {
        float s = 0.0f;
        #pragma unroll
        for (int m = 0; m < MM; ++m) s += fast_exp2(lpv[m][j] - lpmax[j]);
        acc += LN2_F * (lpmax[j] + fast_log2(s));   // back to natural log once
    }

    // ---------- wave32 tree reduce + block reduce + one atomic per block ----------
    #pragma unroll
    for (int off = 16; off > 0; off >>= 1)
        acc += __shfl_down(acc, off, 32);

    const int wid = tid >> 5;
    if ((tid & 31) == 0) s_warp[wid] = acc;
    __syncthreads();
    if (tid == 0) {
        float bs = 0.0f;
        #pragma unroll
        for (int i = 0; i < TPB / 32; ++i) bs += s_warp[i];
        atomicAdd(&out[b], bs);
    }
}

extern "C" void kernel_launch(void* const* d_in, const int* in_sizes, int n_in,
                              void* d_out, int out_size, void* d_ws, size_t ws_size,
                              hipStream_t stream) {
    const float* x           = (const float*)d_in[0];
    const float* logit_probs = (const float*)d_in[1];
    const float* mean        = (const float*)d_in[2];
    const float* log_var     = (const float*)d_in[3];
    const float* coeffs      = (const float*)d_in[4];
    float* out = (float*)d_out;

    zero_out_kernel<<<1, 64, 0, stream>>>(out);

    const long npix    = (long)BB * HWD;          // 524288
    const int  nblocks = (int)(npix / BPIX);      // 512
    dmol_kernel<<<nblocks, TPB, 0, stream>>>(x, logit_probs, mean, log_var, coeffs, out);
}